// RecommendationModel_57801669869915
// MI455X (gfx1250) — compile-verified
//
#include <hip/hip_runtime.h>
#include <math.h>

typedef __attribute__((ext_vector_type(2))) float v2f;
typedef __attribute__((ext_vector_type(8))) float v8f;

#define NROWS   8192
#define DIMIN   256
#define HID     128
#define MARGINF 1.0f
#define BIGV    1e9f
#define HASNEG_THRESH 5e8f

// ---------------------------------------------------------------------------
// MLP layer: Out[8192,128] = act(A[8192,K] @ W[K,128] + bias)
// One wave per 16x16 output tile; 8 waves/block cover the 128 output cols.
// grid.x = 512 row tiles, block = 256 threads.
// ---------------------------------------------------------------------------
__global__ void k_mlp_layer(const float* __restrict__ A, int K,
                            const float* __restrict__ W,
                            const float* __restrict__ bias,
                            float* __restrict__ Out, int do_relu) {
    const int lane    = threadIdx.x & 31;
    const int wave    = threadIdx.x >> 5;     // 0..7 -> column tile
    const int rowBase = blockIdx.x * 16;
    const int colBase = wave * 16;
    const int m   = lane & 15;
    const int hi  = lane >> 4;
    const int klo = hi * 2;

    const float* arow = A + (size_t)(rowBase + m) * K;

    v8f c = {0.f, 0.f, 0.f, 0.f, 0.f, 0.f, 0.f, 0.f};
    for (int k0 = 0; k0 < K; k0 += 4) {
        v2f a = *(const v2f*)(arow + k0 + klo);           // A[m, k0+klo .. +1]
        v2f b;
        b.x = W[(size_t)(k0 + klo)     * HID + colBase + m];
        b.y = W[(size_t)(k0 + klo + 1) * HID + colBase + m];
        c = __builtin_amdgcn_wmma_f32_16x16x4_f32(false, a, false, b,
                                                  (short)0, c, false, false);
    }

    const int col = colBase + m;
    const float bv = bias[col];
#pragma unroll
    for (int r = 0; r < 8; ++r) {
        const int row = rowBase + r + 8 * hi;
        float v = c[r] + bv;
        if (do_relu) v = fmaxf(v, 0.f);
        Out[(size_t)row * HID + col] = v;
    }
}

// ---------------------------------------------------------------------------
// Per-row final score (F @ Wfc + bfc) and squared L2 norm.
// 8 waves/block, one wave per row. grid.x = 1024.
// ---------------------------------------------------------------------------
__global__ void k_score_norm(const float* __restrict__ F,
                             const float* __restrict__ Wfc,
                             const float* __restrict__ bfc,
                             float* __restrict__ scores,
                             float* __restrict__ norms) {
    const int lane = threadIdx.x & 31;
    const int wave = threadIdx.x >> 5;
    const int row  = blockIdx.x * 8 + wave;

    const float4 f = *(const float4*)(F + (size_t)row * HID + lane * 4);
    const float4 w = *(const float4*)(Wfc + lane * 4);
    float dot = f.x * w.x + f.y * w.y + f.z * w.z + f.w * w.w;
    float sq  = f.x * f.x + f.y * f.y + f.z * f.z + f.w * f.w;
#pragma unroll
    for (int off = 16; off; off >>= 1) {
        dot += __shfl_xor(dot, off, 32);
        sq  += __shfl_xor(sq,  off, 32);
    }
    if (lane == 0) {
        scores[row] = dot + bfc[0];
        norms[row]  = sq;
    }
}

// ---------------------------------------------------------------------------
// Pass 1: hardest negative per row.
// grid = (512 row tiles, 4 col chunks), block = 256 (8 waves).
// Each (chunk, wave) worker covers 8 PAIRS of the 256 column-tile pairs.
// A-fragments are hoisted into registers (loop-invariant across col tiles);
// two accumulators -> two independent WMMA dependency chains per wave.
// Row-min combined globally via atomicMin on the float bit pattern
// (valid: all candidates are non-negative floats).
// ---------------------------------------------------------------------------
__global__ void k_hardest_neg(const float* __restrict__ Uf,
                              const float* __restrict__ Mf,
                              const float* __restrict__ u2,
                              const float* __restrict__ v2,
                              const int* __restrict__ labels,
                              unsigned int* __restrict__ hn_bits) {
    const int lane    = threadIdx.x & 31;
    const int wave    = threadIdx.x >> 5;
    const int worker  = blockIdx.y * 8 + wave;   // 0..31
    const int rowBase = blockIdx.x * 16;
    const int m   = lane & 15;
    const int hi  = lane >> 4;
    const int klo = hi * 2;

    int   labRow[8];
    float rn[8];
    float minv[8];
#pragma unroll
    for (int r = 0; r < 8; ++r) {
        const int row = rowBase + r + 8 * hi;
        labRow[r] = labels[row];
        rn[r]     = u2[row];
        minv[r]   = BIGV;
    }

    // Hoist loop-invariant A fragments: 32 x float2 per lane (K = 128).
    const float* arow = Uf + (size_t)(rowBase + m) * HID;
    v2f afr[32];
#pragma unroll
    for (int k = 0; k < 32; ++k) afr[k] = *(const v2f*)(arow + 4 * k + klo);

    for (int p = worker; p < NROWS / 32; p += 32) {     // 256 tile pairs
        const int colBase = p * 32;
        const float* b0row = Mf + (size_t)(colBase + m) * HID;
        const float* b1row = b0row + (size_t)16 * HID;

        v8f c0 = {0.f, 0.f, 0.f, 0.f, 0.f, 0.f, 0.f, 0.f};
        v8f c1 = {0.f, 0.f, 0.f, 0.f, 0.f, 0.f, 0.f, 0.f};
#pragma unroll
        for (int k = 0; k < 32; ++k) {
            v2f b0 = *(const v2f*)(b0row + 4 * k + klo);
            v2f b1 = *(const v2f*)(b1row + 4 * k + klo);
            c0 = __builtin_amdgcn_wmma_f32_16x16x4_f32(false, afr[k], false, b0,
                                                       (short)0, c0, false, false);
            c1 = __builtin_amdgcn_wmma_f32_16x16x4_f32(false, afr[k], false, b1,
                                                       (short)0, c1, false, false);
        }

        const int   col0 = colBase + m;
        const int   col1 = col0 + 16;
        const int   lc0  = labels[col0];
        const int   lc1  = labels[col1];
        const float cn0  = v2[col0];
        const float cn1  = v2[col1];
#pragma unroll
        for (int r = 0; r < 8; ++r) {
            float d0 = sqrtf(fmaxf(rn[r] + cn0 - 2.0f * c0[r], 1e-12f));
            float d1 = sqrtf(fmaxf(rn[r] + cn1 - 2.0f * c1[r], 1e-12f));
            float e0 = (lc0 != labRow[r]) ? d0 : BIGV;
            float e1 = (lc1 != labRow[r]) ? d1 : BIGV;
            minv[r] = fminf(minv[r], fminf(e0, e1));
        }
    }

#pragma unroll
    for (int r = 0; r < 8; ++r) {
        float v = minv[r];
        v = fminf(v, __shfl_xor(v, 1, 32));   // xor 1/2/4/8 reduce within each
        v = fminf(v, __shfl_xor(v, 2, 32));   // 16-lane half independently
        v = fminf(v, __shfl_xor(v, 4, 32));
        v = fminf(v, __shfl_xor(v, 8, 32));
        if (m == 0) {                          // lane 0 -> row r, lane 16 -> row r+8
            atomicMin(hn_bits + rowBase + r + 8 * hi, __float_as_uint(v));
        }
    }
}

// ---------------------------------------------------------------------------
// Pass 2: recompute dist tiles, accumulate hinge sum and valid-pair count.
// Same tiling / hoisting / dual-accumulator structure as pass 1.
// ---------------------------------------------------------------------------
__global__ void k_loss_accum(const float* __restrict__ Uf,
                             const float* __restrict__ Mf,
                             const float* __restrict__ u2,
                             const float* __restrict__ v2,
                             const int* __restrict__ labels,
                             const unsigned int* __restrict__ hn_bits,
                             float* __restrict__ accum_sum,
                             unsigned int* __restrict__ accum_cnt) {
    const int lane    = threadIdx.x & 31;
    const int wave    = threadIdx.x >> 5;
    const int worker  = blockIdx.y * 8 + wave;
    const int rowBase = blockIdx.x * 16;
    const int m   = lane & 15;
    const int hi  = lane >> 4;
    const int klo = hi * 2;

    int   labRow[8];
    float rn[8];
    float hnr[8];
    int   hasneg[8];
#pragma unroll
    for (int r = 0; r < 8; ++r) {
        const int row = rowBase + r + 8 * hi;
        labRow[r] = labels[row];
        rn[r]     = u2[row];
        const float h = __uint_as_float(hn_bits[row]);
        hnr[r]    = h;
        hasneg[r] = (h < HASNEG_THRESH) ? 1 : 0;
    }

    const float* arow = Uf + (size_t)(rowBase + m) * HID;
    v2f afr[32];
#pragma unroll
    for (int k = 0; k < 32; ++k) afr[k] = *(const v2f*)(arow + 4 * k + klo);

    float        fsum = 0.f;
    unsigned int cnt  = 0u;

    for (int p = worker; p < NROWS / 32; p += 32) {
        const int colBase = p * 32;
        const float* b0row = Mf + (size_t)(colBase + m) * HID;
        const float* b1row = b0row + (size_t)16 * HID;

        v8f c0 = {0.f, 0.f, 0.f, 0.f, 0.f, 0.f, 0.f, 0.f};
        v8f c1 = {0.f, 0.f, 0.f, 0.f, 0.f, 0.f, 0.f, 0.f};
#pragma unroll
        for (int k = 0; k < 32; ++k) {
            v2f b0 = *(const v2f*)(b0row + 4 * k + klo);
            v2f b1 = *(const v2f*)(b1row + 4 * k + klo);
            c0 = __builtin_amdgcn_wmma_f32_16x16x4_f32(false, afr[k], false, b0,
                                                       (short)0, c0, false, false);
            c1 = __builtin_amdgcn_wmma_f32_16x16x4_f32(false, afr[k], false, b1,
                                                       (short)0, c1, false, false);
        }

        const int   col0 = colBase + m;
        const int   col1 = col0 + 16;
        const int   lc0  = labels[col0];
        const int   lc1  = labels[col1];
        const float cn0  = v2[col0];
        const float cn1  = v2[col1];
#pragma unroll
        for (int r = 0; r < 8; ++r) {
            if ((lc0 == labRow[r]) && hasneg[r]) {
                const float d = sqrtf(fmaxf(rn[r] + cn0 - 2.0f * c0[r], 1e-12f));
                fsum += fmaxf(d - hnr[r] + MARGINF, 0.f);
                cnt  += 1u;
            }
            if ((lc1 == labRow[r]) && hasneg[r]) {
                const float d = sqrtf(fmaxf(rn[r] + cn1 - 2.0f * c1[r], 1e-12f));
                fsum += fmaxf(d - hnr[r] + MARGINF, 0.f);
                cnt  += 1u;
            }
        }
    }

    float        s = fsum;
    unsigned int k = cnt;
#pragma unroll
    for (int off = 16; off; off >>= 1) {
        s += __shfl_xor(s, off, 32);
        k += (unsigned int)__shfl_xor((int)k, off, 32);
    }
    if (lane == 0) {
        atomicAdd(accum_sum, s);
        atomicAdd(accum_cnt, k);
    }
}

// ---------------------------------------------------------------------------
__global__ void k_init(unsigned int* __restrict__ hn_bits,
                       float* __restrict__ accum_sum,
                       unsigned int* __restrict__ accum_cnt) {
    const int i = blockIdx.x * blockDim.x + threadIdx.x;
    if (i < NROWS) hn_bits[i] = __float_as_uint(BIGV);
    if (i == 0) { *accum_sum = 0.f; *accum_cnt = 0u; }
}

__global__ void k_finalize(const float* __restrict__ accum_sum,
                           const unsigned int* __restrict__ accum_cnt,
                           float* __restrict__ loss_out) {
    const float den = fmaxf((float)(*accum_cnt), 1.0f);
    *loss_out = (*accum_sum) / den;
}

// ---------------------------------------------------------------------------
extern "C" void kernel_launch(void* const* d_in, const int* in_sizes, int n_in,
                              void* d_out, int out_size, void* d_ws, size_t ws_size,
                              hipStream_t stream) {
    (void)in_sizes; (void)n_in; (void)out_size; (void)ws_size;

    const float* user_nodes  = (const float*)d_in[0];   // [8192, 256]
    const float* movie_nodes = (const float*)d_in[1];   // [8192, 256]
    const int*   labels      = (const int*)  d_in[2];   // [8192]
    const float* W1u = (const float*)d_in[3];
    const float* b1u = (const float*)d_in[4];
    const float* W2u = (const float*)d_in[5];
    const float* b2u = (const float*)d_in[6];
    const float* W1m = (const float*)d_in[7];
    const float* b1m = (const float*)d_in[8];
    const float* W2m = (const float*)d_in[9];
    const float* b2m = (const float*)d_in[10];
    const float* Wfc = (const float*)d_in[11];          // [128]
    const float* bfc = (const float*)d_in[12];          // [1]

    float* out = (float*)d_out;          // [0..8191] user, [8192..16383] movie, [16384] loss

    // Workspace layout (floats)
    float* ws  = (float*)d_ws;
    float* Uf  = ws;                               // 8192*128
    float* Mf  = ws + 1 * 1048576;                 // 8192*128
    float* Hb  = ws + 2 * 1048576;                 // 8192*128 (hidden, reused)
    float* u2  = ws + 3 * 1048576;                 // 8192
    float* v2  = u2 + NROWS;                       // 8192
    unsigned int* hn   = (unsigned int*)(v2 + NROWS);   // 8192
    float*        asum = (float*)(hn + NROWS);
    unsigned int* acnt = (unsigned int*)(asum + 1);

    k_init<<<32, 256, 0, stream>>>(hn, asum, acnt);

    // user tower
    k_mlp_layer<<<512, 256, 0, stream>>>(user_nodes, DIMIN, W1u, b1u, Hb, 1);
    k_mlp_layer<<<512, 256, 0, stream>>>(Hb,         HID,   W2u, b2u, Uf, 0);
    // movie tower
    k_mlp_layer<<<512, 256, 0, stream>>>(movie_nodes, DIMIN, W1m, b1m, Hb, 1);
    k_mlp_layer<<<512, 256, 0, stream>>>(Hb,          HID,   W2m, b2m, Mf, 0);

    // scores + squared norms
    k_score_norm<<<1024, 256, 0, stream>>>(Uf, Wfc, bfc, out,          u2);
    k_score_norm<<<1024, 256, 0, stream>>>(Mf, Wfc, bfc, out + NROWS,  v2);

    // pass 1: hardest negative per row
    k_hardest_neg<<<dim3(512, 4), 256, 0, stream>>>(Uf, Mf, u2, v2, labels, hn);
    // pass 2: hinge sum + valid count
    k_loss_accum<<<dim3(512, 4), 256, 0, stream>>>(Uf, Mf, u2, v2, labels, hn,
                                                   asum, acnt);
    // final scalar
    k_finalize<<<1, 1, 0, stream>>>(asum, acnt, out + 2 * NROWS);
}